// TARDIS_72773925863760
// MI455X (gfx1250) — compile-verified
//
#include <hip/hip_runtime.h>
#include <hip/hip_bf16.h>
#include <math.h>

#define Bsz 2048
#define Xs  256
#define Hs  512
#define Rs  64
#define Ms  128
#define EPSg 1e-20f
#define SIG_TAU (10.0f/3.0f)

typedef __attribute__((ext_vector_type(2))) float v2f;
typedef __attribute__((ext_vector_type(8))) float v8f;

__device__ __forceinline__ v8f wmma4(v2f a, v2f b, v8f c) {
  // D = A(16x4,f32) * B(4x16,f32) + C(16x16,f32)
  return __builtin_amdgcn_wmma_f32_16x16x4_f32(false, a, false, b, (short)0, c, false, false);
}
__device__ __forceinline__ float sigmf(float x) { return 1.0f / (1.0f + expf(-x)); }
__device__ __forceinline__ float gumbelf(float u) { return -logf(EPSg - logf(EPSg + u)); }

// ---------------------------------------------------------------------------
// Generic WMMA f32 GEMM:  out[B x N] = concat(A0*s0 | A1*s1 | A2*s2) @ W + bias
// Pair-packed LDS tiles (float2 per lane-fragment -> single ds_load_b64, no
// repack movs; pair-row stride 160 floats = 32 banks so wave halves are
// conflict-free). Double-buffered K pipeline, one barrier per 32-wide chunk.
// BM=64 x BN=64 per block, 256 threads (8 waves), 2 accumulators per wave.
// All segment boundaries are multiples of 32 so each chunk resolves to one
// (source, scale, W-row) uniformly.
// ---------------------------------------------------------------------------
__global__ __launch_bounds__(256) void gemm_cat3(
    const float* __restrict__ s0, int w0, const float* __restrict__ sc0, int wr0,
    const float* __restrict__ s1, int w1, const float* __restrict__ sc1, int wr1,
    const float* __restrict__ s2, int w2p, const float* __restrict__ sc2, int wr2,
    const float* __restrict__ W, int ldw,
    const float* __restrict__ bias,
    float* __restrict__ out, int ldo, int Ktot)
{
  __shared__ v2f a_buf[2][16][80];   // [buf][kpair][m], row stride 160 floats
  __shared__ v2f w_buf[2][16][80];   // [buf][kpair][n]
  const int tid  = threadIdx.x;
  const int gx   = blockIdx.x, gy = blockIdx.y;
  const int lane = tid & 31, wave = tid >> 5;
  const int half = lane >> 4, ml = lane & 15;
  const int m0   = (wave & 3) * 16;
  const int n0   = (wave >> 2) * 32;

  // staging assignments
  const int am     = tid >> 2;         // A row (local) this thread stages
  const int arow   = gy * 64 + am;     // A row (global)
  const int apair0 = (tid & 3) * 4;    // first k-pair (k offset (tid&3)*8)
  const int wp     = tid >> 4;         // W k-pair this thread stages
  const int wn0    = (tid & 15) * 4;   // W col offset

  float areg[8], wreg[8];
  float ascale = 1.0f;

  auto stage_load = [&](int k0) {
    const float* sp; int sw; const float* sc; int wrow; int kl;
    if (k0 < w0)           { sp = s0; sw = w0;  sc = sc0; wrow = wr0 + k0;             kl = k0; }
    else if (k0 < w0 + w1) { sp = s1; sw = w1;  sc = sc1; wrow = wr1 + (k0 - w0);      kl = k0 - w0; }
    else                   { sp = s2; sw = w2p; sc = sc2; wrow = wr2 + (k0 - w0 - w1); kl = k0 - w0 - w1; }
    const float* ap = sp + (size_t)arow * sw + kl + apair0 * 2;
    #pragma unroll
    for (int q = 0; q < 8; ++q) areg[q] = ap[q];
    ascale = sc ? sc[arow] : 1.0f;
    const float* wpp = W + (size_t)(wrow + 2 * wp) * ldw + gx * 64 + wn0;
    #pragma unroll
    for (int q = 0; q < 4; ++q) { wreg[q] = wpp[q]; wreg[4 + q] = wpp[ldw + q]; }
  };
  auto stage_store = [&](int sel) {
    #pragma unroll
    for (int j = 0; j < 4; ++j) {
      v2f t; t.x = areg[2 * j] * ascale; t.y = areg[2 * j + 1] * ascale;
      a_buf[sel][apair0 + j][am] = t;
    }
    #pragma unroll
    for (int j = 0; j < 4; ++j) {
      v2f t; t.x = wreg[j]; t.y = wreg[4 + j];
      w_buf[sel][wp][wn0 + j] = t;
    }
  };

  v8f c0 = {0,0,0,0,0,0,0,0};
  v8f c1 = {0,0,0,0,0,0,0,0};

  const int nch = Ktot / 32;
  stage_load(0);
  stage_store(0);
  __syncthreads();

  for (int ch = 0; ch < nch; ++ch) {
    const int sel = ch & 1;
    if (ch + 1 < nch) stage_load((ch + 1) * 32);   // vmem in flight over wmma
    #pragma unroll
    for (int kk4 = 0; kk4 < 8; ++kk4) {
      int p = kk4 * 2 + half;
      v2f a  = a_buf[sel][p][m0 + ml];
      v2f b0 = w_buf[sel][p][n0 + ml];
      v2f b1 = w_buf[sel][p][n0 + 16 + ml];
      c0 = wmma4(a, b0, c0);
      c1 = wmma4(a, b1, c1);
    }
    if (ch + 1 < nch) stage_store(sel ^ 1);
    __syncthreads();
  }

  #pragma unroll
  for (int i = 0; i < 8; ++i) {
    int row  = gy * 64 + m0 + i + 8 * half;   // C layout: VGPR i -> M=i / M=8+i
    int col0 = gx * 64 + n0 + ml;
    int col1 = col0 + 16;
    out[(size_t)row * ldo + col0] = c0[i] + (bias ? bias[col0] : 0.0f);
    out[(size_t)row * ldo + col1] = c1[i] + (bias ? bias[col1] : 0.0f);
  }
}

// ---------------------------------------------------------------------------
// Routing kernel, one block per batch row:
//  hid = base + keys@Wk + hmem_b @ W_m ; head = sum_k tanh(hid)*vec_a
//  head = l2n(head - 100*prev); y = softmax(head + gumbel); ri = (hard-y)+y
//  r = ri @ hmem_b    (reuses the LDS-resident pair-packed hmem tile)
// ---------------------------------------------------------------------------
__global__ __launch_bounds__(256) void route_kernel(
    const float* __restrict__ hmem, const float* __restrict__ W_fc,
    const float* __restrict__ basep, const float* __restrict__ keysWk,
    const float* __restrict__ vec_a, const float* __restrict__ prev,
    const float* __restrict__ noise, float* __restrict__ ri_out,
    float* __restrict__ r_out)
{
  __shared__ v2f hmp[32][144];   // [rpair][m], row stride 288 floats = 32 banks
  __shared__ v2f wmp[32][144];   // [rpair][k2]
  __shared__ float sm_head[128];
  __shared__ float sm_z[128];
  __shared__ float sm_y[128];
  __shared__ float sm_s[4];
  const int b = blockIdx.x;
  const int tid = threadIdx.x;

  { // stage hmem[b] as pairs: thread -> row m=tid>>1, r-half (tid&1)*32
    const int m = tid >> 1, p0 = (tid & 1) * 16;
    const float* hp = hmem + (size_t)b * Ms * Rs + (size_t)m * 64 + p0 * 2;
    #pragma unroll
    for (int j = 0; j < 16; ++j) {
      v2f t; t.x = hp[2 * j]; t.y = hp[2 * j + 1];
      hmp[p0 + j][m] = t;
    }
  }
  { // stage W_m (W_fc rows 832..895) as pairs: thread -> pair tid>>3, cols (tid&7)*16
    const int p = tid >> 3, k0c = (tid & 7) * 16;
    const float* w0p = W_fc + (size_t)(832 + 2 * p) * 128 + k0c;
    #pragma unroll
    for (int j = 0; j < 16; ++j) {
      v2f t; t.x = w0p[j]; t.y = w0p[128 + j];
      wmp[p][k0c + j] = t;
    }
  }
  __syncthreads();

  const int lane = tid & 31, wave = tid >> 5;
  const int half = lane >> 4, ml = lane & 15;
  const int m0 = wave * 16;   // wave owns rows m0..m0+15, loops all 8 col tiles

  float acc[8];
  #pragma unroll
  for (int i = 0; i < 8; ++i) acc[i] = 0.0f;

  for (int t8 = 0; t8 < 8; ++t8) {
    int n0 = t8 * 16;
    v8f cf = {0,0,0,0,0,0,0,0};
    #pragma unroll
    for (int kk4 = 0; kk4 < 16; ++kk4) {            // K=64, 4 per wmma
      int p = kk4 * 2 + half;
      v2f a  = hmp[p][m0 + ml];
      v2f bb = wmp[p][n0 + ml];
      cf = wmma4(a, bb, cf);
    }
    int k2 = n0 + ml;
    float bk = basep[(size_t)b * 128 + k2];
    float va = vec_a[k2];
    #pragma unroll
    for (int i = 0; i < 8; ++i) {
      int m = m0 + i + 8 * half;
      float v = cf[i] + bk + keysWk[m * 128 + k2];
      acc[i] += tanhf(v) * va;
    }
  }
  // reduce over the 16 columns each half-wave holds
  #pragma unroll
  for (int i = 0; i < 8; ++i) {
    float v = acc[i];
    v += __shfl_xor(v, 1);
    v += __shfl_xor(v, 2);
    v += __shfl_xor(v, 4);
    v += __shfl_xor(v, 8);
    if (ml == 0) sm_head[m0 + i + 8 * half] = v;
  }
  __syncthreads();

  if (tid < 128) sm_z[tid] = sm_head[tid] - 100.0f * prev[(size_t)b * 128 + tid];
  __syncthreads();
  if (tid == 0) {
    float ss = 0.0f;
    for (int m = 0; m < 128; ++m) ss += sm_z[m] * sm_z[m];
    sm_s[0] = fmaxf(sqrtf(ss), 1e-12f);
  }
  __syncthreads();
  if (tid < 128) sm_z[tid] = sm_z[tid] / sm_s[0] + gumbelf(noise[(size_t)b * 128 + tid]); // TAU=1
  __syncthreads();
  if (tid == 0) {
    float mx = -1e30f;
    for (int m = 0; m < 128; ++m) mx = fmaxf(mx, sm_z[m]);
    sm_s[1] = mx;
  }
  __syncthreads();
  if (tid < 128) sm_y[tid] = expf(sm_z[tid] - sm_s[1]);
  __syncthreads();
  if (tid == 0) {
    float s = 0.0f;
    for (int m = 0; m < 128; ++m) s += sm_y[m];
    sm_s[2] = s;
  }
  __syncthreads();
  if (tid < 128) sm_y[tid] = sm_y[tid] / sm_s[2];
  __syncthreads();
  if (tid == 0) {
    float mx = -1.0f;
    for (int m = 0; m < 128; ++m) mx = fmaxf(mx, sm_y[m]);
    sm_s[3] = mx;
  }
  __syncthreads();
  if (tid < 128) {
    float y = sm_y[tid];
    float hard = (y == sm_s[3]) ? 1.0f : 0.0f;
    float rv = (hard - y) + y;   // exactly 0 off-argmax, matches reference fp
    sm_z[tid] = rv;
    ri_out[(size_t)b * 128 + tid] = rv;
  }
  __syncthreads();
  if (tid < 64) { // r = ri @ hmem_b from the LDS pair tile
    int p = tid >> 1, odd = tid & 1;
    float s = 0.0f;
    for (int m = 0; m < 128; ++m) {
      v2f t = hmp[p][m];
      s += sm_z[m] * (odd ? t.y : t.x);
    }
    r_out[(size_t)b * 64 + tid] = s;
  }
}

// --------------------------- small helper kernels ---------------------------
__global__ void keyswk_kernel(const float* __restrict__ keys,
                              const float* __restrict__ W_fc,
                              float* __restrict__ out) {
  int m = blockIdx.x, k2 = threadIdx.x;
  float s = 0.0f;
  for (int r = 0; r < 64; ++r) s += keys[m * 64 + r] * W_fc[(size_t)(768 + r) * 128 + k2];
  out[m * 128 + k2] = s;
}

__global__ void un_kernel(const float* __restrict__ u_t, float* __restrict__ un) {
  __shared__ float s[128];
  __shared__ float nrm;
  int b = blockIdx.x, t = threadIdx.x;
  float u = u_t[(size_t)b * 128 + t];
  s[t] = u * u;
  __syncthreads();
  if (t == 0) {
    float ss = 0.0f;
    for (int i = 0; i < 128; ++i) ss += s[i];
    nrm = fmaxf(sqrtf(ss), 1e-12f);
  }
  __syncthreads();
  un[(size_t)b * 128 + t] = u / nrm;
}

__global__ void ab_kernel(const float* __restrict__ x, const float* __restrict__ h,
                          const float* __restrict__ r, const float* __restrict__ W1,
                          const float* __restrict__ b1, const float* __restrict__ noise,
                          float* __restrict__ alpha, float* __restrict__ beta) {
  __shared__ float s0[128], s1[128];
  int b = blockIdx.x, t = threadIdx.x;
  float a0 = 0.0f, a1 = 0.0f;
  for (int k = t; k < 832; k += 128) {
    float v;
    if (k < 256)      v = x[(size_t)b * 256 + k];
    else if (k < 768) v = h[(size_t)b * 512 + (k - 256)];
    else              v = r[(size_t)b * 64 + (k - 768)];
    a0 += v * W1[k * 2];
    a1 += v * W1[k * 2 + 1];
  }
  s0[t] = a0; s1[t] = a1;
  __syncthreads();
  if (t == 0) {
    float t0 = 0.0f, t1 = 0.0f;
    for (int i = 0; i < 128; ++i) { t0 += s0[i]; t1 += s1[i]; }
    alpha[b] = sigmf((t0 + b1[0] + gumbelf(noise[(size_t)b * 2 + 0])) * SIG_TAU);
    beta[b]  = sigmf((t1 + b1[1] + gumbelf(noise[(size_t)b * 2 + 1])) * SIG_TAU);
  }
}

__global__ void point_kernel(const float* __restrict__ gates, const float* __restrict__ w2v,
                             const float* __restrict__ c, const float* __restrict__ r,
                             float* __restrict__ out, float* __restrict__ newh) {
  int idx = blockIdx.x * blockDim.x + threadIdx.x;
  if (idx >= Bsz * 576) return;
  int b = idx / 576, n = idx % 576;
  if (n < 512) {
    float ig = gates[(size_t)b * 1536 + n];
    float fg = gates[(size_t)b * 1536 + 512 + n];
    float og = gates[(size_t)b * 1536 + 1024 + n];
    float nc = c[(size_t)b * 512 + n] * sigmf(fg + 1.0f)
             + sigmf(ig) * tanhf(w2v[(size_t)b * 512 + n]);
    float nh = tanhf(nc) * sigmf(og);
    out[(size_t)b * 576 + n] = nh;
    newh[(size_t)b * 512 + n] = nh;
  } else {
    out[(size_t)b * 576 + n] = r[(size_t)b * 64 + (n - 512)];
  }
}

__global__ void hmem_kernel(const float* __restrict__ hmem, const float* __restrict__ hw,
                            const float* __restrict__ ri, const int* __restrict__ memcnt,
                            float* __restrict__ out2) {
  const float4* hm4 = (const float4*)hmem;
  const float4* hw4 = (const float4*)hw;
  float4* o4 = (float4*)out2;
  int mc = memcnt[0];
  const int total = Bsz * Ms * 16;   // float4 elements
  for (int idx = blockIdx.x * blockDim.x + threadIdx.x; idx < total;
       idx += gridDim.x * blockDim.x) {
    int b   = idx / (Ms * 16);
    int rem = idx - b * (Ms * 16);
    int m   = rem >> 4;
    int jj  = rem & 15;
    float wi  = (mc < Ms) ? ((m == mc) ? 1.0f : 0.0f) : ri[(size_t)b * Ms + m];
    float omw = 1.0f - wi;
    float4 hv = hm4[idx];
    float4 wv = hw4[b * 16 + jj];
    float4 o;
    o.x = wv.x * wi + hv.x * omw;
    o.y = wv.y * wi + hv.y * omw;
    o.z = wv.z * wi + hv.z * omw;
    o.w = wv.w * wi + hv.w * omw;
    o4[idx] = o;
  }
}

// ---------------------------------------------------------------------------
extern "C" void kernel_launch(void* const* d_in, const int* in_sizes, int n_in,
                              void* d_out, int out_size, void* d_ws, size_t ws_size,
                              hipStream_t stream) {
  (void)in_sizes; (void)n_in; (void)out_size; (void)ws_size;
  const float* x        = (const float*)d_in[0];
  const float* h        = (const float*)d_in[1];
  const float* c        = (const float*)d_in[2];
  const float* hmem     = (const float*)d_in[3];
  const float* u_t      = (const float*)d_in[4];
  const float* prev     = (const float*)d_in[5];
  const float* W_full   = (const float*)d_in[6];
  const float* bias     = (const float*)d_in[7];
  const float* W_full1  = (const float*)d_in[8];
  const float* bias1    = (const float*)d_in[9];
  const float* W_full2  = (const float*)d_in[10];
  const float* bias2    = (const float*)d_in[11];
  const float* keys     = (const float*)d_in[12];
  const float* vec_a    = (const float*)d_in[13];
  const float* W_fc     = (const float*)d_in[14];
  const float* b_fc     = (const float*)d_in[15];
  const float* W_fc1    = (const float*)d_in[16];
  const float* b_fc1    = (const float*)d_in[17];
  const float* noise_sm = (const float*)d_in[18];
  const float* noise_sg = (const float*)d_in[19];
  const int*   memcnt   = (const int*)d_in[20];

  float* out      = (float*)d_out;
  float* out_hmem = out + (size_t)Bsz * 576;

  float* ws     = (float*)d_ws;
  float* keysWk = ws;  ws += 128 * 128;
  float* un     = ws;  ws += (size_t)Bsz * 128;
  float* basep  = ws;  ws += (size_t)Bsz * 128;
  float* ri     = ws;  ws += (size_t)Bsz * 128;
  float* rbuf   = ws;  ws += (size_t)Bsz * 64;
  float* gates  = ws;  ws += (size_t)Bsz * 1536;
  float* alpha  = ws;  ws += Bsz;
  float* beta   = ws;  ws += Bsz;
  float* w2v    = ws;  ws += (size_t)Bsz * 512;
  float* newh   = ws;  ws += (size_t)Bsz * 512;
  float* hw     = ws;  ws += (size_t)Bsz * 64;

  keyswk_kernel<<<128, 128, 0, stream>>>(keys, W_fc, keysWk);
  un_kernel<<<Bsz, 128, 0, stream>>>(u_t, un);
  // base = [x|c|un] @ W_fc rows {0..767, 896..1023} + b_fc
  gemm_cat3<<<dim3(2, Bsz / 64), 256, 0, stream>>>(
      x, 256, nullptr, 0,  c, 512, nullptr, 256,  un, 128, nullptr, 896,
      W_fc, 128, b_fc, basep, 128, 896);
  route_kernel<<<Bsz, 256, 0, stream>>>(hmem, W_fc, basep, keysWk, vec_a,
                                        prev, noise_sm, ri, rbuf);
  // gates = [x|h|r] @ W_full + bias
  gemm_cat3<<<dim3(24, Bsz / 64), 256, 0, stream>>>(
      x, 256, nullptr, 0,  h, 512, nullptr, 256,  rbuf, 64, nullptr, 768,
      W_full, 1536, bias, gates, 1536, 832);
  ab_kernel<<<Bsz, 128, 0, stream>>>(x, h, rbuf, W_full1, bias1, noise_sg, alpha, beta);
  // w2v = [x | alpha*h | beta*r] @ W_full2 + bias2 (row scales folded into A staging)
  gemm_cat3<<<dim3(8, Bsz / 64), 256, 0, stream>>>(
      x, 256, nullptr, 0,  h, 512, alpha, 256,  rbuf, 64, beta, 768,
      W_full2, 512, bias2, w2v, 512, 832);
  point_kernel<<<(Bsz * 576 + 255) / 256, 256, 0, stream>>>(gates, w2v, c, rbuf, out, newh);
  // hw = [x|new_h] @ W_fc1 + b_fc1
  gemm_cat3<<<dim3(1, Bsz / 64), 256, 0, stream>>>(
      x, 256, nullptr, 0,  newh, 512, nullptr, 256,  nullptr, 1, nullptr, 0,
      W_fc1, 64, b_fc1, hw, 64, 768);
  hmem_kernel<<<2048, 256, 0, stream>>>(hmem, hw, ri, memcnt, out_hmem);
}